// CTCLossLayer_66769561583923
// MI455X (gfx1250) — compile-verified
//
#include <hip/hip_runtime.h>
#include <stdint.h>

// Problem constants (from reference): B=4096, T=256, C=96, L=16, S=33
#define CT_T     256
#define CT_C     96
#define CT_L     16
#define NEGV     (-1.0e30f)
#define ROWP     100            // padded LDS row stride in floats (bank-conflict-free)
#define TILE_T   16
#define NTILES   (CT_T / TILE_T)
#define LN2      0.6931471805599453f
#define LOG2E    1.4426950408889634f

typedef float v2f __attribute__((ext_vector_type(2)));
typedef float v8f __attribute__((ext_vector_type(8)));

__device__ __forceinline__ float log2v(float x) { return __builtin_amdgcn_logf(x); }   // v_log_f32
__device__ __forceinline__ float exp2v(float x) { return __builtin_amdgcn_exp2f(x); }  // v_exp_f32

__device__ __forceinline__ float rfl_f(float x) {
    return __uint_as_float(__builtin_amdgcn_readfirstlane(__float_as_uint(x)));
}
__device__ __forceinline__ float rdl31_f(float x) {
    return __uint_as_float(__builtin_amdgcn_readlane(__float_as_uint(x), 31));
}

// logaddexp via raw v_exp/v_log (inputs never denormal/inf; -1e30 sentinel safe)
__device__ __forceinline__ float lae(float x, float y) {
    float m = fmaxf(x, y);
    float d = fabsf(x - y);
    return m + LN2 * log2v(1.0f + exp2v(-d * LOG2E));
}

// Issue one 16x96-float tile as 12 per-lane async b128 copies global->LDS,
// scattering into the padded (ROWP=100) LDS layout.
__device__ __forceinline__ void issue_tile_async(const float* __restrict__ gsrc,
                                                 float* buf, int lane) {
#pragma unroll
    for (int i = 0; i < 12; ++i) {
        int idx = i * 32 + lane;          // 0..383 float4 units in the tile
        int tl  = idx / 24;               // timestep within tile (24 float4 per row)
        int c4  = idx - tl * 24;          // float4 index within row
        const float* g = gsrc + tl * CT_C + c4 * 4;
        uint32_t lA = (uint32_t)(uintptr_t)(buf + tl * ROWP + c4 * 4);
        asm volatile("global_load_async_to_lds_b128 %0, %1, off"
                     :: "v"(lA), "v"(g) : "memory");
    }
}

__global__ __launch_bounds__(32) void ctc_fwd_kernel(
        const float* __restrict__ y,          // [B,T,C]
        const int*   __restrict__ labels,     // [B,L]
        const int*   __restrict__ input_len,  // [B,1]
        const int*   __restrict__ label_len,  // [B,1]
        float* __restrict__ out)              // [B,1]
{
    __shared__ float lds[2 * TILE_T * ROWP + TILE_T];   // double buffer + 16 tile sums
    float* const sums = lds + 2 * TILE_T * ROWP;
    const int b    = blockIdx.x;
    const int lane = threadIdx.x;

    const int il = __builtin_amdgcn_readfirstlane(input_len[b]);
    const int ll = __builtin_amdgcn_readfirstlane(label_len[b]);

    // ---- extended label sequence: ext[s] = blank (even s) / labels[(s-1)/2] (odd s)
    int labv = labels[b * CT_L + (lane & 15)];
    int srcC = (lane >= 1) ? ((lane - 1) >> 1) : 0;
    int srcP = (lane >= 3) ? ((lane - 3) >> 1) : 0;
    int labC = __shfl(labv, srcC, 32);
    int labP = __shfl(labv, srcP, 32);
    const int  myCls  = (lane & 1) ? labC : (CT_C - 1);
    const bool skipok = (lane & 1) && (lane >= 3) && (labC != labP);
    const bool sval   = lane < (2 * ll + 1);
    const bool v32    = 32 < (2 * ll + 1);

    // WMMA f32 16x16x4 A-fragment mapping: lanes 0..15 hold K0,K1; 16..31 hold K2,K3
    const int aM = lane & 15;
    const int k0 = (lane >> 4) * 2;

    const float* yb = y + (size_t)b * CT_T * CT_C;

    // prefetch tile 0
    issue_tile_async(yb, lds, lane);

    float a = NEGV, a32 = NEGV;           // alpha[lane], alpha[32] (replicated)
    const v2f bones = {1.0f, 1.0f};

    for (int tt = 0; tt < NTILES; ++tt) {
        float* buf = lds + (tt & 1) * (TILE_T * ROWP);
        if (tt + 1 < NTILES) {
            issue_tile_async(yb + (size_t)(tt + 1) * TILE_T * CT_C,
                             lds + ((tt + 1) & 1) * (TILE_T * ROWP), lane);
            asm volatile("s_wait_asynccnt 12" ::: "memory"); // current buffer done
        } else {
            asm volatile("s_wait_asynccnt 0" ::: "memory");
        }

        // ---- per-timestep channel sums via one V_WMMA_F32_16X16X4_F32
        // A[m,k] = sum_j y[t=m, 4j+k]; B = ones(4x16)  =>  D[m,*] = sum_c y[m,c]
        v2f acc = {0.0f, 0.0f};
#pragma unroll
        for (int j = 0; j < 24; ++j) {
            const float2 v = *(const float2*)(buf + aM * ROWP + 4 * j + k0);
            acc.x += v.x; acc.y += v.y;
        }
        v8f csum = {};
        csum = __builtin_amdgcn_wmma_f32_16x16x4_f32(
            false, acc, false, bones, (short)0, csum, false, false);

        // D row m lives in vgpr m (lanes 0-15) / m-8 (lanes 16-31); park sums in LDS
        if ((lane & 15) == 0) {
            const int base = (lane >> 4) * 8;
#pragma unroll
            for (int v = 0; v < 8; ++v) sums[base + v] = csum[v];
        }

        // ---- DP over the 16 timesteps of this tile
#pragma unroll 4
        for (int tl = 0; tl < TILE_T; ++tl) {
            float sumv = sums[tl];                         // LDS broadcast read
            float l2Z  = log2v(sumv + (float)CT_C * 1e-7f);
            float yv   = buf[tl * ROWP + myCls];
            float e    = LN2 * (log2v(yv + 1e-7f) - l2Z);  // emit[t, s=lane]
            float e32  = rfl_f(e);                         // lane 0 = blank emit
            if (tt == 0 && tl == 0) {
                a   = (lane == 0 || (lane == 1 && ll >= 1)) ? e : NEGV;
                a32 = NEGV;
            } else {
                float p1  = (lane >= 1) ? __shfl_up(a, 1, 32) : NEGV;
                float p2  = skipok      ? __shfl_up(a, 2, 32) : NEGV;
                float a31 = rdl31_f(a);
                float na  = lae(lae(a, p1), p2) + e;
                na = sval ? na : NEGV;
                float n32 = lae(a32, a31) + e32;
                n32 = v32 ? n32 : NEGV;
                const bool upd = (tt * TILE_T + tl) < il;
                a   = upd ? na  : a;
                a32 = upd ? n32 : a32;
            }
        }
    }

    // ---- final readout
    int twoLL = 2 * ll;
    int liB = (twoLL < 32) ? twoLL : 0;
    float ab_lane = __shfl(a, liB, 32);
    float a_blank = (twoLL >= 32) ? a32 : ab_lane;
    int liL = twoLL - 1; if (liL < 0) liL = 0; if (liL > 31) liL = 31;
    float a_lab = __shfl(a, liL, 32);
    float logp = (ll > 0) ? lae(a_blank, a_lab) : a_blank;
    if (lane == 0) out[b] = -logp;
}

extern "C" void kernel_launch(void* const* d_in, const int* in_sizes, int n_in,
                              void* d_out, int out_size, void* d_ws, size_t ws_size,
                              hipStream_t stream) {
    const float* y      = (const float*)d_in[0];
    const int*   labels = (const int*)d_in[1];
    const int*   il     = (const int*)d_in[2];
    const int*   ll     = (const int*)d_in[3];
    float*       out    = (float*)d_out;
    const int Bn = in_sizes[2];   // input_length has B elements
    ctc_fwd_kernel<<<Bn, 32, 0, stream>>>(y, labels, il, ll, out);
}